// SoftmaxTopK_44848048505290
// MI455X (gfx1250) — compile-verified
//
#include <hip/hip_runtime.h>
#include <hip/hip_bf16.h>
#include <stdint.h>

// ---------------------------------------------------------------------------
// MoE gating: softmax over E=64 experts + top-K=8 per token (1M tokens).
// Memory floor: 320 MB @ 23.3 TB/s ~ 14 us; selection VALU is the real limit,
// so top-8 uses sorted 8-chunks + bitonic top-8 merges instead of insertion.
// CDNA5 path: async global->LDS staging (ASYNCcnt) + wave32 layout.
// ---------------------------------------------------------------------------

#define E   64          // experts per token
#define K   8           // top-k (reference top_k == 8; d_in[1] ignored)
#define TPB 128         // 4 wave32s; one token per thread
#define ROW_STRIDE 66   // floats per LDS row: 8B-aligned, bank-conflict-free

typedef unsigned long long u64;

#define HAVE_ASYNC_LDS 0
#if defined(__has_builtin)
#if __has_builtin(__builtin_amdgcn_global_load_async_to_lds_b64)
#undef HAVE_ASYNC_LDS
#define HAVE_ASYNC_LDS 1
#endif
#if __has_builtin(__builtin_amdgcn_s_wait_asynccnt)
#define HAVE_WAIT_ASYNC 1
#endif
#if __has_builtin(__builtin_amdgcn_exp2f)
#define EXP2F(v) __builtin_amdgcn_exp2f(v)
#endif
#endif
#ifndef EXP2F
#define EXP2F(v) exp2f(v)
#endif

#define LOG2E 1.44269504088896340736f

#if HAVE_ASYNC_LDS
// Builtin signature (from clang diag): (v2i AS(1)*, v2i AS(3)*, imm, imm).
typedef int v2i_async __attribute__((vector_size(8)));
typedef __attribute__((address_space(1))) v2i_async* gbl_v2i_ptr;
typedef __attribute__((address_space(3))) v2i_async* lds_v2i_ptr;
#endif

__device__ __forceinline__ void stage_pair(const float* gp, float* lp) {
#if HAVE_ASYNC_LDS
  __builtin_amdgcn_global_load_async_to_lds_b64(
      (gbl_v2i_ptr)(uintptr_t)gp, (lds_v2i_ptr)(uintptr_t)lp,
      /*offset=*/0, /*cpol=*/0);
#else
  float2 v = *(const float2*)gp;
  lp[0] = v.x;
  lp[1] = v.y;
#endif
}

// ---- exact-order keys: u64 = mono32(x) << 6 | (63 - idx) -------------------
// mono32 preserves float total order; low bits make keys distinct and encode
// the jax tie-break (equal value -> lower index ranks first under descending).
__device__ __forceinline__ u64 make_key(float x, unsigned idx) {
  unsigned b = __float_as_uint(x);
  unsigned m = b ^ ((unsigned)((int)b >> 31) | 0x80000000u);
  return ((u64)m << 6) | (u64)(63u - idx);
}
__device__ __forceinline__ float key_value(u64 k) {
  unsigned m = (unsigned)(k >> 6);
  unsigned b = (m & 0x80000000u) ? (m ^ 0x80000000u) : ~m;
  return __uint_as_float(b);
}
__device__ __forceinline__ unsigned key_index(u64 k) {
  return 63u - ((unsigned)k & 63u);
}

// Compare-exchange: a := max, b := min  (v_cmp_gt_u64 + 4 cndmask)
__device__ __forceinline__ void ce(u64& a, u64& b) {
  u64 hi = a < b ? b : a;
  u64 lo = a < b ? a : b;
  a = hi;
  b = lo;
}
__device__ __forceinline__ u64 umax64(u64 a, u64 b) { return a < b ? b : a; }

// Optimal 19-CE network, descending order.
__device__ __forceinline__ void sort8(u64 (&v)[8]) {
  ce(v[0], v[1]); ce(v[2], v[3]); ce(v[4], v[5]); ce(v[6], v[7]);
  ce(v[0], v[2]); ce(v[1], v[3]); ce(v[4], v[6]); ce(v[5], v[7]);
  ce(v[1], v[2]); ce(v[5], v[6]); ce(v[0], v[4]); ce(v[3], v[7]);
  ce(v[1], v[5]); ce(v[2], v[6]);
  ce(v[1], v[4]); ce(v[3], v[6]);
  ce(v[2], v[4]); ce(v[3], v[5]);
  ce(v[3], v[4]);
}

// Descending bitonic cleanup of 8 (12 CE).
__device__ __forceinline__ void bitonic8(u64 (&v)[8]) {
  ce(v[0], v[4]); ce(v[1], v[5]); ce(v[2], v[6]); ce(v[3], v[7]);
  ce(v[0], v[2]); ce(v[1], v[3]); ce(v[4], v[6]); ce(v[5], v[7]);
  ce(v[0], v[1]); ce(v[2], v[3]); ce(v[4], v[5]); ce(v[6], v[7]);
}

// Load 8 consecutive floats of the row as keys (chunk k -> elements 8k..8k+7).
__device__ __forceinline__ void load_chunk(const float2* row2, int k, u64 (&ck)[8]) {
#pragma unroll
  for (int q = 0; q < 4; ++q) {
    float2 v = row2[4 * k + q];
    ck[2 * q]     = make_key(v.x, (unsigned)(8 * k + 2 * q));
    ck[2 * q + 1] = make_key(v.y, (unsigned)(8 * k + 2 * q + 1));
  }
}

__global__ __launch_bounds__(TPB)
void SoftmaxTopK_44848048505290_kernel(const float* __restrict__ x,
                                       float* __restrict__ out_vals,
                                       float* __restrict__ out_idx,
                                       int ntok) {
  __shared__ float tile[TPB * ROW_STRIDE];   // 128 * 66 * 4 = 33,792 B

  const int tid      = threadIdx.x;
  const int blk_tok0 = blockIdx.x * TPB;

  // ---- Stage 128 rows x 64 f32 into LDS (async 8B per lane per op) ----
  const int    CHUNKS = TPB * (E / 2);       // 4096 x 8B; block source is linear
  const float* gbase  = x + (size_t)blk_tok0 * E;
  if (blk_tok0 + TPB <= ntok) {              // uniform fast path: full block
#pragma unroll 4
    for (int c = tid; c < CHUNKS; c += TPB) {
      stage_pair(gbase + 2 * c, &tile[(c >> 5) * ROW_STRIDE + (c & 31) * 2]);
    }
  } else {                                    // ragged tail block
#pragma unroll 4
    for (int c = tid; c < CHUNKS; c += TPB) {
      if (blk_tok0 + (c >> 5) < ntok) {
        stage_pair(gbase + 2 * c, &tile[(c >> 5) * ROW_STRIDE + (c & 31) * 2]);
      }
    }
  }
#if HAVE_ASYNC_LDS
#if defined(HAVE_WAIT_ASYNC)
  __builtin_amdgcn_s_wait_asynccnt(0);
#else
  asm volatile("s_wait_asynccnt 0" ::: "memory");
#endif
#endif
  __syncthreads();

  const int tok = blk_tok0 + tid;
  if (tok >= ntok) return;                   // no barriers after this point

  const float2* row2 = (const float2*)&tile[tid * ROW_STRIDE];

  // ---- Top-8 selection: sort chunk0, then fold 7 sorted chunks ----
  u64 R[8];
  load_chunk(row2, 0, R);
  sort8(R);
#pragma unroll
  for (int k = 1; k < 8; ++k) {
    u64 ck[8];
    load_chunk(row2, k, ck);
    sort8(ck);
    // top-8 of two descending 8-lists: max(R[i], ck[7-i]) is the top-8
    // multiset and is bitonic -> 12-CE cleanup re-sorts it.
    u64 c[8];
#pragma unroll
    for (int i = 0; i < 8; ++i) c[i] = umax64(R[i], ck[7 - i]);
    bitonic8(c);
#pragma unroll
    for (int i = 0; i < 8; ++i) R[i] = c[i];
  }

  const float mx = key_value(R[0]);          // row max = top-1, exact

  // ---- softmax denominator ----
  float s = 0.0f;
#pragma unroll 8
  for (int i = 0; i < E / 2; ++i) {
    float2 v = row2[i];
    s += EXP2F((v.x - mx) * LOG2E);
    s += EXP2F((v.y - mx) * LOG2E);
  }
  const float inv = 1.0f / s;

  // ---- emit top-8 probs + indices (exact values recovered from keys) ----
  float pv[K], pi[K];
#pragma unroll
  for (int j = 0; j < K; ++j) {
    pv[j] = EXP2F((key_value(R[j]) - mx) * LOG2E) * inv;
    pi[j] = (float)key_index(R[j]);
  }

  float4* vdst = (float4*)(out_vals + (size_t)tok * K);
  vdst[0] = make_float4(pv[0], pv[1], pv[2], pv[3]);
  vdst[1] = make_float4(pv[4], pv[5], pv[6], pv[7]);
  float4* idst = (float4*)(out_idx + (size_t)tok * K);
  idst[0] = make_float4(pi[0], pi[1], pi[2], pi[3]);
  idst[1] = make_float4(pi[4], pi[5], pi[6], pi[7]);
}

extern "C" void kernel_launch(void* const* d_in, const int* in_sizes, int n_in,
                              void* d_out, int out_size, void* d_ws, size_t ws_size,
                              hipStream_t stream) {
  (void)n_in; (void)d_ws; (void)ws_size; (void)out_size;
  const float* x = (const float*)d_in[0];
  // d_in[1] is top_k == 8 (compile-time K); selection depth is hardcoded.
  const int ntok = in_sizes[0] / E;

  float* out  = (float*)d_out;
  float* vals = out;                         // [ntok, 8] f32, descending
  float* idxs = out + (size_t)ntok * K;      // [ntok, 8] indices as f32

  const int blocks = (ntok + TPB - 1) / TPB;
  hipLaunchKernelGGL(SoftmaxTopK_44848048505290_kernel,
                     dim3(blocks), dim3(TPB), 0, stream,
                     x, vals, idxs, ntok);
}